// RefractiveInterface_3195455668404
// MI455X (gfx1250) — compile-verified
//
#include <hip/hip_runtime.h>
#include <hip/hip_bf16.h>
#include <stdint.h>

// ---------------------------------------------------------------------------
// RefractiveInterface forward projection for MI455X (gfx1250, wave32).
// Per-ray quartic solve (Ferrari + Cardano) in fp64, selection in fp32.
// Data movement: TDM (tensor_load_to_lds) double-buffered tile streaming of
// pos3d into LDS, overlapped with the fp64 compute; synced via TENSORcnt.
// ---------------------------------------------------------------------------

#define TILE 256              // rays per tile == blockDim.x
#define TILE_ELEMS (TILE * 3) // floats per tile

#if defined(__has_builtin)
#if __has_builtin(__builtin_amdgcn_tensor_load_to_lds)
#define HAVE_TDM 1
#else
#define HAVE_TDM 0
#endif
#else
#define HAVE_TDM 0
#endif

#if defined(__has_builtin) && __has_builtin(__builtin_amdgcn_s_wait_tensorcnt)
#define WAIT_TENSORCNT(n) __builtin_amdgcn_s_wait_tensorcnt((short)(n))
#else
#define WAIT_TENSORCNT(n) asm volatile("s_wait_tensorcnt %0" ::"i"(n) : "memory")
#endif

typedef unsigned int u32x4 __attribute__((ext_vector_type(4)));
typedef int i32x4 __attribute__((ext_vector_type(4)));
typedef int i32x8 __attribute__((ext_vector_type(8)));

// ---------------------------------------------------------------------------
// Per-ray math. Mirrors the reference numerically:
//   coefficients fp32 -> quartic solve fp64 -> root selection fp32.
// ---------------------------------------------------------------------------
__device__ __forceinline__ void compute_ray(float px, float py, float pz,
                                            float z1x, float z1y, float z1z,
                                            float d, float* __restrict__ o) {
    const float mu = 1.33f;
    const float mu2 = mu * mu;
    const float k = mu2 - 1.0f; // == c1

    // por = cross(normal, p), normal = -z1 ; z2 = cross(por, z1) = cross(z1, c_hat)
    float cx = z1y * pz - z1z * py;
    float cy = z1z * px - z1x * pz;
    float cz = z1x * py - z1y * px;
    float rn = rsqrtf(cx * cx + cy * cy + cz * cz);
    cx *= rn; cy *= rn; cz *= rn;
    float z2x = z1y * cz - z1z * cy;
    float z2y = z1z * cx - z1x * cz;
    float z2z = z1x * cy - z1y * cx;
    float v = px * z1x + py * z1y + pz * z1z;
    float u = px * z2x + py * z2y + pz * z2z;

    // Quartic coefficients in fp32 (as reference), c1 == k.
    float c2 = -2.0f * u * k;
    float c3 = k * (d * d + u * u) + 2.0f * d * v - v * v;
    float c4 = -2.0f * d * d * mu2 * u;
    float c5 = d * d * mu2 * u * u;

    // --- fp64 Ferrari -------------------------------------------------------
    double A3 = (double)c2 / (double)k;
    double A2 = (double)c3 / (double)k;
    double A1 = (double)c4 / (double)k;
    double A0 = (double)c5 / (double)k;

    double p = A2 - 0.375 * A3 * A3;
    double q = A1 - 0.5 * A2 * A3 + 0.125 * A3 * A3 * A3;
    double A34 = 0.25 * A3;
    double r = A0 - A1 * A34 + A2 * A34 * A34 - 3.0 * A34 * A34 * A34 * A34;

    // Resolvent cubic: y^3 + B2 y^2 + B1 y + B0
    double B2 = 2.0 * p;
    double B1 = p * p - 4.0 * r;
    double B0 = -(q * q);
    double p3 = B1 - B2 * B2 * (1.0 / 3.0);
    double q3 = B0 - B1 * B2 * (1.0 / 3.0) + B2 * B2 * B2 * (2.0 / 27.0);
    double Dd = 0.25 * q3 * q3 + p3 * p3 * p3 * (1.0 / 27.0) + 1e-12;

    double tdep;
    if (Dd >= 0.0) {
        double s = sqrt(Dd);
        tdep = cbrt(-0.5 * q3 + s) + cbrt(-0.5 * q3 - s); // unique real root
    } else {
        // three real roots: take the largest (what the reference's principal-
        // branch Cardano x0 evaluates to in this regime)
        double m = sqrt(-p3 * (1.0 / 3.0));
        double arg = (-0.5 * q3) / (m * m * m);
        arg = fmin(1.0, fmax(-1.0, arg));
        tdep = 2.0 * m * cos(acos(arg) * (1.0 / 3.0));
    }
    double tt = tdep - B2 * (1.0 / 3.0);
    tt = fmax(tt, 1e-6);

    double st = sqrt(tt);
    double hq = st * q / (2.0 * tt);
    double ph = 0.5 * (p + tt);
    double ca = ph - hq;
    double cb = ph + hq;
    double d1 = tt - 4.0 * ca; // disc of y^2 + st y + ca
    double d2 = tt - 4.0 * cb; // disc of y^2 - st y + cb
    double s1 = sqrt(fmax(d1, 0.0));
    double s2 = sqrt(fmax(d2, 0.0));
    // |imag| > 1e-3 <=> disc < -4e-6  -> invalid candidate
    bool ok1 = d1 > -4e-6;
    bool ok2 = d2 > -4e-6;

    float xs0 = (float)(0.5 * (-st + s1) - A34);
    float xs1 = (float)(0.5 * (-st - s1) - A34);
    float xs2 = (float)(0.5 * (st + s2) - A34);
    float xs3 = (float)(0.5 * (st - s2) - A34);

    // --- fp32 root selection by refraction error ---------------------------
    const float inv_mu = 1.0f / mu;
    float xs[4] = {xs0, xs1, xs2, xs3};
    bool ok[4] = {ok1, ok1, ok2, ok2};
    float best = xs[0];
    float berr = __builtin_inff();
#pragma unroll
    for (int i = 0; i < 4; ++i) {
        float x = xs[i];
        float e = __builtin_inff();
        if (ok[i]) {
            float bbb = x * x + d * d;
            float sq = sqrtf(fmaf(k, bbb, d * d)); // d^2 - (1-mu^2)*bbb
            float b = (d - sq) * inv_mu;
            float vrx = x * inv_mu;
            float vry = d * inv_mu - b;
            e = fabsf((u - x) * vry - (v - d) * vrx);
        }
        if (e < berr) { berr = e; best = x; }
    }

    o[0] = best * z2x + d * z1x;
    o[1] = best * z2y + d * z1y;
    o[2] = best * z2z + d * z1z;
}

#if HAVE_TDM
// ---------------------------------------------------------------------------
// TDM: issue a 1-D tile load (nElem fp32 elements, tile_dim0 = 768) into LDS.
// D# group0: count=1, lds_addr, global_addr(57b), type=2 ("image").
// D# group1: workgroup_mask=0, data_size=2 (4 bytes), tensor_dim0=nElem
//            (tail tiles read beyond -> zeros), tile_dim0=768, stride=768.
// Groups 2/3 zero (dims 2..4 unused).
// ---------------------------------------------------------------------------
__device__ __forceinline__ void tdm_load_tile(const float* gsrc,
                                              unsigned lds_off,
                                              unsigned nElem) {
    unsigned long long ga = (unsigned long long)(uintptr_t)gsrc;
    u32x4 g0;
    g0[0] = 1u;                 // count=1, is_restore=0, gather off
    g0[1] = lds_off;            // LDS byte address
    g0[2] = (unsigned)ga;       // global_addr[31:0]
    g0[3] = (unsigned)((ga >> 32) & 0x1FFFFFFu) | 0x80000000u; // [56:32] | type=2

    i32x8 g1;
    g1[0] = (int)(2u << 16);                       // mask=0, data_size=4B
    g1[1] = (int)((nElem & 0xFFFFu) << 16);        // tensor_dim0[15:0]
    g1[2] = (int)((nElem >> 16) | (1u << 16));     // tensor_dim0[31:16] | tensor_dim1=1
    g1[3] = (int)((unsigned)TILE_ELEMS << 16);     // tile_dim0 = 768
    g1[4] = 0;                                     // tile_dim1=0, tile_dim2=0 (unused)
    g1[5] = (int)TILE_ELEMS;                       // tensor_dim0_stride lo
    g1[6] = 0;
    g1[7] = 0;

    i32x4 z4 = {0, 0, 0, 0};
#if defined(__clang_major__) && (__clang_major__ >= 23)
    i32x8 z8 = {0, 0, 0, 0, 0, 0, 0, 0};
    __builtin_amdgcn_tensor_load_to_lds(g0, g1, z4, z4, z8, 0);
#else
    __builtin_amdgcn_tensor_load_to_lds(g0, g1, z4, z4, 0);
#endif
}

__global__ void __launch_bounds__(TILE)
refract_tdm_kernel(const float* __restrict__ pos3d,
                   const float* __restrict__ nvec,
                   const float* __restrict__ dscal,
                   float* __restrict__ out,
                   int nrays, int ntiles) {
    __shared__ float tile_buf[2][TILE_ELEMS];

    // Uniform setup (scalar-cached loads; negligible).
    float n0 = nvec[0], n1 = nvec[1];
    float d = fmaxf(dscal[0], 0.0f) + 0.001f; // relu(d)/SCALE + 0.001
    float inv_nn = rsqrtf(n0 * n0 + n1 * n1 + 1.0f);
    float z1x = n0 * inv_nn, z1y = n1 * inv_nn, z1z = inv_nn;

    const int tid = threadIdx.x;
    const int tstep = gridDim.x;
    const bool leader = (tid < 32); // wave 0 owns the TDM pipe

    long long total = 3LL * (long long)nrays;
    int t0 = blockIdx.x;
    int cur = 0;

    if (t0 < ntiles && leader) {
        long long e0 = (long long)t0 * TILE_ELEMS;
        unsigned rem = (unsigned)min((long long)TILE_ELEMS, total - e0);
        tdm_load_tile(pos3d + e0, (unsigned)(uintptr_t)&tile_buf[0][0], rem);
    }

    for (int t = t0; t < ntiles; t += tstep) {
        int tn = t + tstep;
        if (leader) {
            if (tn < ntiles) { // prefetch next tile into the other buffer
                long long en = (long long)tn * TILE_ELEMS;
                unsigned rem = (unsigned)min((long long)TILE_ELEMS, total - en);
                tdm_load_tile(pos3d + en,
                              (unsigned)(uintptr_t)&tile_buf[cur ^ 1][0], rem);
                WAIT_TENSORCNT(1); // current tile's DMA complete (in-order)
            } else {
                WAIT_TENSORCNT(0);
            }
        }
        __syncthreads(); // tile_buf[cur] visible to all 8 waves

        int ray = t * TILE + tid;
        if (ray < nrays) {
            float px = tile_buf[cur][3 * tid + 0];
            float py = tile_buf[cur][3 * tid + 1];
            float pz = tile_buf[cur][3 * tid + 2];
            compute_ray(px, py, pz, z1x, z1y, z1z, d, out + 3 * (size_t)ray);
        }
        __syncthreads(); // everyone done reading before next DMA overwrites
        cur ^= 1;
    }
}
#else
// Fallback: direct global loads (toolchain without the TDM builtin).
__global__ void __launch_bounds__(TILE)
refract_plain_kernel(const float* __restrict__ pos3d,
                     const float* __restrict__ nvec,
                     const float* __restrict__ dscal,
                     float* __restrict__ out, int nrays) {
    float n0 = nvec[0], n1 = nvec[1];
    float d = fmaxf(dscal[0], 0.0f) + 0.001f;
    float inv_nn = rsqrtf(n0 * n0 + n1 * n1 + 1.0f);
    float z1x = n0 * inv_nn, z1y = n1 * inv_nn, z1z = inv_nn;

    int stride = gridDim.x * blockDim.x;
    for (int ray = blockIdx.x * blockDim.x + threadIdx.x; ray < nrays;
         ray += stride) {
        const float* p = pos3d + 3 * (size_t)ray;
        __builtin_prefetch(p + 3 * (size_t)stride, 0, 1); // global_prefetch_b8
        compute_ray(p[0], p[1], p[2], z1x, z1y, z1z, d, out + 3 * (size_t)ray);
    }
}
#endif

extern "C" void kernel_launch(void* const* d_in, const int* in_sizes, int n_in,
                              void* d_out, int out_size, void* d_ws,
                              size_t ws_size, hipStream_t stream) {
    const float* pos3d = (const float*)d_in[0]; // [N,3] f32
    const float* nvec = (const float*)d_in[1];  // [2]   f32
    const float* dsc = (const float*)d_in[2];   // [1]   f32
    float* out = (float*)d_out;                 // [N,3] f32

    int nrays = in_sizes[0] / 3;
    int ntiles = (nrays + TILE - 1) / TILE;

#if HAVE_TDM
    // Persistent blocks so each runs several tiles -> real DMA/compute overlap.
    int blocks = ntiles < 2048 ? ntiles : 2048;
    refract_tdm_kernel<<<blocks, TILE, 0, stream>>>(pos3d, nvec, dsc, out,
                                                    nrays, ntiles);
#else
    int blocks = ntiles < 4096 ? ntiles : 4096;
    refract_plain_kernel<<<blocks, TILE, 0, stream>>>(pos3d, nvec, dsc, out,
                                                      nrays);
#endif
    (void)d_ws; (void)ws_size; (void)n_in; (void)out_size;
}